// MPNN_49916109914129
// MI455X (gfx1250) — compile-verified
//
#include <hip/hip_runtime.h>

// MPNN forward for MI455X (gfx1250, wave32).
//
// Phase 1 (k_gemm3):   msg1/msg2/h1 = features @ {w1,w2,wo} + bias, via
//                      V_WMMA_F32_16X16X4_F32 (fp32-exact matrix pipe).
// Phase 2 (k_maxpart): partial S[slice,b,j,m] = max_{i in slice: adj[b,i,j]!=0} msg2[b,i,m]
//                      4 sender-slices x 64 j-tiles x 2 batches = 512 blocks for occupancy;
//                      LDS-tiled (32 senders x 16 receivers) + global_prefetch of next tile.
//          (k_comb):   xr = relu(msg1 + max of the 4 partials)   [exact: max associative]
// Phase 3 (k_mlp):     4-layer MLP on xr via WMMA with LDS activation staging,
//                      final layer fused with + h1, written to d_out.

typedef float v2f __attribute__((ext_vector_type(2)));
typedef float v8f __attribute__((ext_vector_type(8)));

#define DD     128
#define NN     1024
#define BB     2
#define MROWS  (BB * NN)        // 2048 total rows
#define SLAB   262144           // 2048*128 floats per ws buffer

// ---- fp32 WMMA 16x16x4 fragment helpers (layouts per CDNA5 ISA 7.12.2) ----

// A (16x4, MxK): lanes 0-15 -> M=lane, {v0,v1}={K0,K1}; lanes 16-31 -> {K2,K3}
static __device__ __forceinline__ v2f ld_a(const float* base, int ld, int m0, int k0, int lane) {
    int row = m0 + (lane & 15);
    int k   = k0 + ((lane >> 4) << 1);
    const float* p = base + (size_t)row * ld + k;
    v2f r; r.x = p[0]; r.y = p[1]; return r;
}
// B (4x16, KxN): lanes 0-15 -> N=lane, {v0,v1}={K0,K1}; lanes 16-31 -> {K2,K3}
static __device__ __forceinline__ v2f ld_b(const float* w, int ld, int k0, int n0, int lane) {
    int k   = k0 + ((lane >> 4) << 1);
    int col = n0 + (lane & 15);
    v2f r; r.x = w[(size_t)k * ld + col]; r.y = w[(size_t)(k + 1) * ld + col]; return r;
}

static __device__ __forceinline__ v8f wmma4(v2f a, v2f b, v8f c) {
    // (neg_a, A, neg_b, B, c_mod, C, reuse_a, reuse_b)
    return __builtin_amdgcn_wmma_f32_16x16x4_f32(false, a, false, b, (short)0, c, false, false);
}

// ---------------- Phase 1: three 2048x128 @ 128x128 GEMMs ----------------
// 1024 output tiles (16x16) per matrix, 3 matrices = 3072 wave-tiles.
// 384 blocks x 8 waves: every wave owns exactly one tile.
__global__ __launch_bounds__(256) void k_gemm3(
    const float* __restrict__ feat,
    const float* __restrict__ w1, const float* __restrict__ b1,
    const float* __restrict__ w2, const float* __restrict__ b2,
    const float* __restrict__ wo, const float* __restrict__ bo,
    float* __restrict__ msg1, float* __restrict__ msg2, float* __restrict__ h1)
{
    int lane  = threadIdx.x & 31;
    int t     = blockIdx.x * 8 + (threadIdx.x >> 5);
    int which = t >> 10;            // 0:w1 1:w2 2:wo
    int tloc  = t & 1023;
    int m0    = (tloc >> 3) << 4;   // row tile (2048/16 = 128 tiles)
    int n0    = (tloc & 7) << 4;    // col tile (128/16 = 8 tiles)

    const float* W  = (which == 0) ? w1 : (which == 1) ? w2 : wo;
    const float* Bv = (which == 0) ? b1 : (which == 1) ? b2 : bo;
    float*       O  = (which == 0) ? msg1 : (which == 1) ? msg2 : h1;

    v8f c = {};
#pragma unroll 8
    for (int k0 = 0; k0 < DD; k0 += 4) {
        v2f a = ld_a(feat, DD, m0, k0, lane);
        v2f b = ld_b(W, DD, k0, n0, lane);
        c = wmma4(a, b, c);
    }
    int col   = n0 + (lane & 15);
    float bias = Bv[col];
    int rbase = m0 + ((lane >> 4) << 3);   // C: VGPR r -> M = r + 8*(lane>=16)
#pragma unroll
    for (int r = 0; r < 8; ++r)
        O[(size_t)(rbase + r) * DD + col] = c[r] + bias;
}

// ---------------- Phase 2a: partial masked max over sender slices ----------------
// Block = (b, i-slice of 256 senders, 16 receiver columns j).
// Thread (jl, 8-channel chunk m0). Stage 32 sender rows of msg2 (16KB) + 32x16
// adj block (2KB) in LDS; prefetch next msg2 tile while the VALU loop runs.
__global__ __launch_bounds__(256) void k_maxpart(
    const float* __restrict__ msg2, const float* __restrict__ adj,
    float* __restrict__ part)   // 4 slabs of 2048x128 partial maxima
{
    __shared__ float sM[32 * 128];
    __shared__ float sA[32 * 16];

    int b     = blockIdx.x >> 8;
    int rem   = blockIdx.x & 255;
    int slice = rem >> 6;               // 0..3 -> senders [slice*256, slice*256+256)
    int j0    = (rem & 63) << 4;
    int t     = threadIdx.x;
    int jl    = t >> 4;
    int m0    = (t & 15) << 3;

    const float NEG = -__builtin_inff();
    float S[8];
#pragma unroll
    for (int q = 0; q < 8; ++q) S[q] = NEG;

    const float* m2b  = msg2 + (size_t)b * NN * DD;
    const float* adjb = adj  + (size_t)b * NN * NN;

    int iBeg = slice << 8, iEnd = iBeg + 256;
    for (int i0 = iBeg; i0 < iEnd; i0 += 32) {
        // stage 32x128 msg2 chunk (1024 float4, 4 per thread)
        const float4* src = (const float4*)(m2b + (size_t)i0 * DD);
        float4* dst = (float4*)sM;
#pragma unroll
        for (int q = 0; q < 4; ++q) dst[q * 256 + t] = src[q * 256 + t];
        // stage 32x16 adj block (512 floats, 2 per thread)
#pragma unroll
        for (int q = 0; q < 2; ++q) {
            int e = q * 256 + t;
            sA[e] = adjb[(size_t)(i0 + (e >> 4)) * NN + j0 + (e & 15)];
        }
        // pull next 16KB msg2 tile toward the caches while we compute this one
        if (i0 + 32 < iEnd)
            __builtin_prefetch((const char*)(m2b + (size_t)(i0 + 32) * DD) + t * 64, 0, 3);
        __syncthreads();
#pragma unroll 4
        for (int ii = 0; ii < 32; ++ii) {
            float a = sA[ii * 16 + jl];
            const float* mr = &sM[ii * 128 + m0];
            float4 u = *(const float4*)mr;
            float4 v = *(const float4*)(mr + 4);
            float vals[8] = {u.x, u.y, u.z, u.w, v.x, v.y, v.z, v.w};
            bool on = (a != 0.0f);
#pragma unroll
            for (int q = 0; q < 8; ++q)
                S[q] = fmaxf(S[q], on ? vals[q] : NEG);
        }
        __syncthreads();
    }
    int j = j0 + jl;
    float* pr = part + (size_t)slice * SLAB + ((size_t)b * NN + j) * DD + m0;
#pragma unroll
    for (int q = 0; q < 8; ++q) pr[q] = S[q];
}

// ---------------- Phase 2b: combine 4 partials, add msg1, relu ----------------
__global__ __launch_bounds__(256) void k_comb(
    const float* __restrict__ msg1, const float* __restrict__ part,
    float* __restrict__ xr)
{
    int e = blockIdx.x * 256 + threadIdx.x;          // float4 index, 65536 total
    const float4* p0 = (const float4*)part;
    const float4* p1 = (const float4*)(part + SLAB);
    const float4* p2 = (const float4*)(part + 2 * SLAB);
    const float4* p3 = (const float4*)(part + 3 * SLAB);
    float4 a = p0[e], b = p1[e], c = p2[e], d = p3[e];
    float4 m = ((const float4*)msg1)[e];
    float4 r;
    r.x = fmaxf(m.x + fmaxf(fmaxf(a.x, b.x), fmaxf(c.x, d.x)), 0.0f);
    r.y = fmaxf(m.y + fmaxf(fmaxf(a.y, b.y), fmaxf(c.y, d.y)), 0.0f);
    r.z = fmaxf(m.z + fmaxf(fmaxf(a.z, b.z), fmaxf(c.z, d.z)), 0.0f);
    r.w = fmaxf(m.w + fmaxf(fmaxf(a.w, b.w), fmaxf(c.w, d.w)), 0.0f);
    ((float4*)xr)[e] = r;                            // == relu(masked-max) exactly
}

// ---------------- Phase 3: WMMA MLP 128->32->64->64->128, + h1 ----------------
// Block of 8 waves handles 32 rows; activations ping-pong through LDS so the
// C-matrix (row-across-lanes) layout can be re-read as A-matrix fragments.
__global__ __launch_bounds__(256) void k_mlp(
    const float* __restrict__ x, const float* __restrict__ h1,
    const float* __restrict__ wm0, const float* __restrict__ bm0,
    const float* __restrict__ wm1, const float* __restrict__ bm1,
    const float* __restrict__ wm2, const float* __restrict__ bm2,
    const float* __restrict__ wm3, const float* __restrict__ bm3,
    float* __restrict__ out)
{
    __shared__ float s1[32 * 32];
    __shared__ float s2[32 * 64];
    __shared__ float s3[32 * 64];

    int lane = threadIdx.x & 31;
    int wv   = threadIdx.x >> 5;
    int row0 = blockIdx.x << 5;
    int cl   = lane & 15;
    int hb   = (lane >> 4) << 3;   // C row offset: +8 for upper half-wave

    // layer 0: x(32x128) @ wm0(128x32), 4 tile-jobs (waves 0..3)
    if (wv < 4) {
        int m0 = (wv >> 1) << 4, n0 = (wv & 1) << 4;
        v8f c = {};
#pragma unroll 8
        for (int k0 = 0; k0 < 128; k0 += 4)
            c = wmma4(ld_a(x + (size_t)row0 * 128, 128, m0, k0, lane),
                      ld_b(wm0, 32, k0, n0, lane), c);
        int col = n0 + cl; float bias = bm0[col]; int rb = m0 + hb;
#pragma unroll
        for (int r = 0; r < 8; ++r)
            s1[(rb + r) * 32 + col] = fmaxf(c[r] + bias, 0.0f);
    }
    __syncthreads();

    // layer 1: s1(32x32) @ wm1(32x64), 8 jobs (one per wave)
    {
        int m0 = (wv >> 2) << 4, n0 = (wv & 3) << 4;
        v8f c = {};
#pragma unroll
        for (int k0 = 0; k0 < 32; k0 += 4)
            c = wmma4(ld_a(s1, 32, m0, k0, lane), ld_b(wm1, 64, k0, n0, lane), c);
        int col = n0 + cl; float bias = bm1[col]; int rb = m0 + hb;
#pragma unroll
        for (int r = 0; r < 8; ++r)
            s2[(rb + r) * 64 + col] = fmaxf(c[r] + bias, 0.0f);
    }
    __syncthreads();

    // layer 2: s2(32x64) @ wm2(64x64), 8 jobs
    {
        int m0 = (wv >> 2) << 4, n0 = (wv & 3) << 4;
        v8f c = {};
#pragma unroll
        for (int k0 = 0; k0 < 64; k0 += 4)
            c = wmma4(ld_a(s2, 64, m0, k0, lane), ld_b(wm2, 64, k0, n0, lane), c);
        int col = n0 + cl; float bias = bm2[col]; int rb = m0 + hb;
#pragma unroll
        for (int r = 0; r < 8; ++r)
            s3[(rb + r) * 64 + col] = fmaxf(c[r] + bias, 0.0f);
    }
    __syncthreads();

    // layer 3: s3(32x64) @ wm3(64x128), 16 jobs (2 per wave); fuse + h1 -> out
#pragma unroll
    for (int job = wv; job < 16; job += 8) {
        int m0 = (job >> 3) << 4, n0 = (job & 7) << 4;
        v8f c = {};
#pragma unroll
        for (int k0 = 0; k0 < 64; k0 += 4)
            c = wmma4(ld_a(s3, 64, m0, k0, lane), ld_b(wm3, 128, k0, n0, lane), c);
        int col = n0 + cl; float bias = bm3[col]; int rb = m0 + hb;
#pragma unroll
        for (int r = 0; r < 8; ++r) {
            size_t gr = (size_t)(row0 + rb + r) * 128 + col;
            out[gr] = c[r] + bias + h1[gr];
        }
    }
}

extern "C" void kernel_launch(void* const* d_in, const int* in_sizes, int n_in,
                              void* d_out, int out_size, void* d_ws, size_t ws_size,
                              hipStream_t stream) {
    (void)in_sizes; (void)n_in; (void)out_size; (void)ws_size;
    const float* feat = (const float*)d_in[0];
    // d_in[1] e_features, d_in[2] g_features: shape-checked but unused by reference
    const float* adj  = (const float*)d_in[3];
    const float* w1   = (const float*)d_in[4];  const float* b1 = (const float*)d_in[5];
    const float* w2   = (const float*)d_in[6];  const float* b2 = (const float*)d_in[7];
    const float* wo   = (const float*)d_in[8];  const float* bo = (const float*)d_in[9];
    const float* wm0  = (const float*)d_in[10]; const float* bm0 = (const float*)d_in[11];
    const float* wm1  = (const float*)d_in[12]; const float* bm1 = (const float*)d_in[13];
    const float* wm2  = (const float*)d_in[14]; const float* bm2 = (const float*)d_in[15];
    const float* wm3  = (const float*)d_in[16]; const float* bm3 = (const float*)d_in[17];
    float* out = (float*)d_out;

    float* ws   = (float*)d_ws;          // needs 8 * 2048*128 floats = 8 MB
    float* msg1 = ws;
    float* msg2 = ws + SLAB;
    float* h1   = ws + 2 * SLAB;
    float* xr   = ws + 3 * SLAB;
    float* part = ws + 4 * SLAB;         // 4 partial-max slabs

    k_gemm3  <<<384, 256, 0, stream>>>(feat, w1, b1, w2, b2, wo, bo, msg1, msg2, h1);
    k_maxpart<<<512, 256, 0, stream>>>(msg2, adj, part);
    k_comb   <<<256, 256, 0, stream>>>(msg1, part, xr);
    k_mlp    <<< 64, 256, 0, stream>>>(xr, h1, wm0, bm0, wm1, bm1, wm2, bm2, wm3, bm3, out);
}